// ALIGNN_18837726560686
// MI455X (gfx1250) — compile-verified
//
#include <hip/hip_runtime.h>

// Problem sizes (fixed by the reference)
#define Nn 10000
#define Ne 160000
#define Nt 480000
#define NBg 16
#define HIDD 256

typedef float v2f __attribute__((ext_vector_type(2)));
typedef float v8f __attribute__((ext_vector_type(8)));

// ---------------------------------------------------------------------------
// GEMM: C[M x Nc] = A[M x K] @ W[K x Nc] + bias (+ C if acc)
// fp32 WMMA 16x16x4. Block = 128 threads (4 waves), tile 64(M) x BN(N), BK=16.
// BNF = fragments per wave in N (4 -> BN=64, 8 -> BN=128).
// Both A and B tiles are stored K-contiguous so every fragment is one
// 2-dword LDS load (no cross-row shuffles before the WMMA).
// ---------------------------------------------------------------------------
template<int BNF>
__global__ __launch_bounds__(128) void k_gemm_wmma(
    const float* __restrict__ A, const float* __restrict__ Wm,
    const float* __restrict__ bias, float* __restrict__ C,
    int M, int K, int Nc, int acc)
{
  constexpr int BN = BNF * 16;
  __shared__ float As[64][17];    // [row][k], +1 pad: conflict-free frag reads
  __shared__ float BsT[BN][17];   // [col][k], K-contiguous like As
  const int tid  = threadIdx.x;
  const int m0   = blockIdx.x * 64;
  const int n0   = blockIdx.y * BN;
  const int wave = tid >> 5;
  const int lane = tid & 31;
  const int lh   = lane & 15;          // row/col within half
  const int kh   = (lane >> 4) << 1;   // 0 (lanes 0-15) or 2 (lanes 16-31)

  v8f zf = {0.f,0.f,0.f,0.f,0.f,0.f,0.f,0.f};
  v8f cf[BNF];
#pragma unroll
  for (int j = 0; j < BNF; ++j) cf[j] = zf;

  const int ak = tid & 15, ar0 = tid >> 4;   // A tile loader coords

  for (int k0 = 0; k0 < K; k0 += 16) {
#pragma unroll
    for (int i = 0; i < 8; ++i) {
      int r = ar0 + i * 8;
      int gr = m0 + r, gk = k0 + ak;
      As[r][ak] = (gr < M && gk < K) ? A[(long long)gr * K + gk] : 0.f;
    }
    // speculative prefetch of the next A K-chunk (global_prefetch_b8)
    if (k0 + 16 < K) {
      int gr = m0 + ar0;
      if (gr < M) __builtin_prefetch(&A[(long long)gr * K + (k0 + 16 + ak)], 0, 1);
    }
#pragma unroll
    for (int idx = 0; idx < 2 * BNF; ++idx) {
      int e   = tid + idx * 128;
      int col = e % BN;
      int kk  = e / BN;
      int gk  = k0 + kk;
      BsT[col][kk] = (gk < K) ? Wm[(long long)gk * Nc + (n0 + col)] : 0.f;
    }
    __syncthreads();
#pragma unroll
    for (int k4 = 0; k4 < 16; k4 += 4) {
      v2f a;
      const float* ap = &As[wave * 16 + lh][k4 + kh];
      a.x = ap[0];
      a.y = ap[1];
#pragma unroll
      for (int j = 0; j < BNF; ++j) {
        v2f b;
        const float* bp = &BsT[j * 16 + lh][k4 + kh];
        b.x = bp[0];
        b.y = bp[1];
        // 8 args: (neg_a, A, neg_b, B, c_mod, C, reuse_a, reuse_b)
        cf[j] = __builtin_amdgcn_wmma_f32_16x16x4_f32(
            false, a, false, b, (short)0, cf[j], false, false);
      }
    }
    __syncthreads();
  }
  // C/D layout: VGPR i -> row i (lanes 0-15) / row i+8 (lanes 16-31)
  const int rbase = m0 + wave * 16 + ((lane >> 4) << 3);
#pragma unroll
  for (int j = 0; j < BNF; ++j) {
    int col = n0 + j * 16 + lh;
    float bv = bias ? bias[col] : 0.f;
#pragma unroll
    for (int i = 0; i < 8; ++i) {
      int row = rbase + i;
      if (row < M) {
        long long idx = (long long)row * Nc + col;
        float v = cf[j][i] + bv;
        if (acc) v += C[idx];
        C[idx] = v;
      }
    }
  }
}

// ---------------------------------------------------------------------------
// Elementwise / reduction helpers
// ---------------------------------------------------------------------------
__global__ void k_zero(float* __restrict__ p, long long n)
{
  long long t = (long long)blockIdx.x * blockDim.x + threadIdx.x;
  if (t < n) p[t] = 0.f;
}

// partial column sums/sumsq (blockDim.x == C, one block covers 512 rows)
__global__ void k_colstat(const float* __restrict__ X, long long R, int C,
                          float* __restrict__ sum, float* __restrict__ sumsq)
{
  int c = threadIdx.x;
  long long r0 = (long long)blockIdx.x * 512;
  long long r1 = r0 + 512; if (r1 > R) r1 = R;
  float s = 0.f, s2 = 0.f;
  for (long long r = r0; r < r1; ++r) {
    float v = X[r * C + c];
    s += v; s2 += v * v;
  }
  atomicAdd(&sum[c], s);
  atomicAdd(&sumsq[c], s2);
}

__global__ void k_statfin(float* __restrict__ st, int C, float invR)
{
  int c = threadIdx.x;
  float mean = st[c] * invR;
  float var  = st[C + c] * invR - mean * mean;   // biased var (ddof=0)
  st[2 * C + c] = mean;
  st[3 * C + c] = rsqrtf(var + 1e-5f);           // EPS_BN
}

// out = silu(g*(x-mean)*inv + b)    (add=1: out += ...)
__global__ void k_bnsilu(const float* __restrict__ X,
                         const float* __restrict__ mean,
                         const float* __restrict__ inv,
                         const float* __restrict__ g,
                         const float* __restrict__ b,
                         float* __restrict__ OUT,
                         long long n, int cmask, int add)
{
  long long t = (long long)blockIdx.x * blockDim.x + threadIdx.x;
  if (t >= n) return;
  int c = (int)t & cmask;
  float u = g[c] * (X[t] - mean[c]) * inv[c] + b[c];
  float a = u / (1.f + expf(-u));
  if (add) OUT[t] += a; else OUT[t] = a;
}

// per-edge message + sigmoid + float-atomic segment sums
__global__ void k_scatter(const float4* __restrict__ sg, const float4* __restrict__ dg,
                          const float4* __restrict__ bh, float4* __restrict__ msg,
                          float* __restrict__ ssh, float* __restrict__ ss,
                          const int* __restrict__ si, const int* __restrict__ di,
                          long long nE)
{
  long long t = (long long)blockIdx.x * blockDim.x + threadIdx.x;
  if (t >= nE * 64) return;
  long long e = t >> 6;
  int q = (int)(t & 63);
  int s = si[e], d = di[e];
  float4 m = msg[e * 64 + q];             // holds edge_gate(e)+bias
  float4 a = sg[(long long)s * 64 + q];
  float4 b = dg[(long long)d * 64 + q];
  m.x += a.x + b.x; m.y += a.y + b.y; m.z += a.z + b.z; m.w += a.w + b.w;
  msg[e * 64 + q] = m;
  float4 hb = bh[(long long)s * 64 + q];
  float s0 = 1.f / (1.f + expf(-m.x));
  float s1 = 1.f / (1.f + expf(-m.y));
  float s2 = 1.f / (1.f + expf(-m.z));
  float s3 = 1.f / (1.f + expf(-m.w));
  long long base = (long long)d * 256 + q * 4;
  atomicAdd(&ssh[base + 0], s0 * hb.x);
  atomicAdd(&ssh[base + 1], s1 * hb.y);
  atomicAdd(&ssh[base + 2], s2 * hb.z);
  atomicAdd(&ssh[base + 3], s3 * hb.w);
  atomicAdd(&ss[base + 0], s0);
  atomicAdd(&ss[base + 1], s1);
  atomicAdd(&ss[base + 2], s2);
  atomicAdd(&ss[base + 3], s3);
}

__global__ void k_hdiv(float* __restrict__ ssh, const float* __restrict__ ss, long long n)
{
  long long t = (long long)blockIdx.x * blockDim.x + threadIdx.x;
  if (t < n) ssh[t] = ssh[t] / (ss[t] + 1e-6f);   // EPS_AGG
}

__global__ void k_poolacc(const float4* __restrict__ x, const int* __restrict__ n2g,
                          float* __restrict__ gf, float* __restrict__ cnt, int n)
{
  long long t = (long long)blockIdx.x * blockDim.x + threadIdx.x;
  if (t >= (long long)n * 64) return;
  long long i = t >> 6; int q = (int)(t & 63);
  int g = n2g[i];
  float4 v = x[i * 64 + q];
  long long base = (long long)g * 256 + q * 4;
  atomicAdd(&gf[base + 0], v.x);
  atomicAdd(&gf[base + 1], v.y);
  atomicAdd(&gf[base + 2], v.z);
  atomicAdd(&gf[base + 3], v.w);
  if (q == 0) atomicAdd(&cnt[g], 1.f);
}

__global__ void k_poolfin(float* __restrict__ gf, const float* __restrict__ cnt, int nb)
{
  int t = blockIdx.x * blockDim.x + threadIdx.x;
  if (t < nb * 256) gf[t] /= fmaxf(cnt[t >> 8], 1.f);
}

// seq-len-1 cross attention collapses: ctx_g = g2t_v(text).
// eg = LN(gfeat + graph_out(ctx_g)); out = eg @ w_fc + b_fc
__global__ void k_finale(const float* __restrict__ gfeat, const float* __restrict__ text,
                         const float* __restrict__ wv, const float* __restrict__ bv,   // g2t_v 64x256
                         const float* __restrict__ wo, const float* __restrict__ bo,   // graph_out 256x256
                         const float* __restrict__ lg, const float* __restrict__ lb,   // ln_g
                         const float* __restrict__ wfc, const float* __restrict__ bfc, // fc 256x1, 1
                         float* __restrict__ out)
{
  __shared__ float tv[256];
  __shared__ float red[256];
  int b = blockIdx.x, t = threadIdx.x;
  float a = bv[t];
  for (int k = 0; k < 64; ++k) a += text[b * 64 + k] * wv[k * 256 + t];
  tv[t] = a;
  __syncthreads();
  float go = bo[t];
  for (int k = 0; k < 256; ++k) go += tv[k] * wo[k * 256 + t];
  float pre = gfeat[b * 256 + t] + go;
  red[t] = pre; __syncthreads();
  for (int s = 128; s > 0; s >>= 1) { if (t < s) red[t] += red[t + s]; __syncthreads(); }
  float mean = red[0] * (1.f / 256.f);
  __syncthreads();
  float dv = pre - mean;
  red[t] = dv * dv; __syncthreads();
  for (int s = 128; s > 0; s >>= 1) { if (t < s) red[t] += red[t + s]; __syncthreads(); }
  float var = red[0] * (1.f / 256.f);
  __syncthreads();
  float eg = lg[t] * dv * rsqrtf(var + 1e-5f) + lb[t];
  red[t] = eg * wfc[t]; __syncthreads();
  for (int s = 128; s > 0; s >>= 1) { if (t < s) red[t] += red[t + s]; __syncthreads(); }
  if (t == 0) out[b] = red[0] + bfc[0];
}

// ---------------------------------------------------------------------------
// Host orchestration
// ---------------------------------------------------------------------------
static inline unsigned cdivu(long long a, long long b) { return (unsigned)((a + b - 1) / b); }

static inline void h_gemm(const float* A, const float* Wm, const float* bias, float* C,
                          int M, int K, int Nc, int acc, hipStream_t s)
{
  if ((Nc & 127) == 0) {
    dim3 g(cdivu(M, 64), Nc / 128), b(128);
    k_gemm_wmma<8><<<g, b, 0, s>>>(A, Wm, bias, C, M, K, Nc, acc);
  } else {
    dim3 g(cdivu(M, 64), Nc / 64), b(128);
    k_gemm_wmma<4><<<g, b, 0, s>>>(A, Wm, bias, C, M, K, Nc, acc);
  }
}

static inline void h_zero(float* p, long long n, hipStream_t s)
{
  k_zero<<<cdivu(n, 256), 256, 0, s>>>(p, n);
}

static inline void h_stats(const float* X, long long R, int C, float* st, hipStream_t s)
{
  h_zero(st, 2LL * C, s);
  k_colstat<<<cdivu(R, 512), C, 0, s>>>(X, R, C, st, st + C);
  k_statfin<<<1, C, 0, s>>>(st, C, 1.f / (float)R);
}

static inline void h_bnsilu(const float* X, const float* st, int C,
                            const float* g, const float* b, float* OUT,
                            long long R, int add, hipStream_t s)
{
  long long n = R * C;
  k_bnsilu<<<cdivu(n, 256), 256, 0, s>>>(X, st + 2 * C, st + 3 * C, g, b, OUT, n, C - 1, add);
}

extern "C" void kernel_launch(void* const* d_in, const int* in_sizes, int n_in,
                              void* d_out, int out_size, void* d_ws, size_t ws_size,
                              hipStream_t stream)
{
  (void)in_sizes; (void)n_in; (void)out_size; (void)ws_size;
  auto P  = [&](int i) -> const float* { return (const float*)d_in[i]; };
  auto Ii = [&](int i) -> const int*   { return (const int*)d_in[i]; };

  const float* x_atom = P(0);
  const float* e_edge = P(1);
  const float* t_trip = P(2);
  const float* text   = P(3);
  const int* src    = Ii(4);
  const int* dst    = Ii(5);
  const int* lg_src = Ii(6);
  const int* lg_dst = Ii(7);
  const int* n2g    = Ii(8);

  // ---- param leaf indices (jax tree flatten: dict keys sorted) ----
  // alignn starts at 9: per layer {edge(14), node(14)}; eggc leaf order:
  //   bnE.b, bnE.g, bnN.b, bnN.g, dg.b, dg.w, du.b, du.w, eg.b, eg.w, sg.b, sg.w, su.b, su.w
  const int L_ALIGNN = 9;                 // 4*28 = 112 leaves
  const int L_ANGLE  = 121;               // 2 mlps * 4 = 8
  const int L_ATOM   = 129;               // 4
  const int L_ATTN   = 133;               // 20
  const int L_EDGEE  = 153;               // 8
  const int L_FC     = 161;               // 2
  const int L_GCN    = 163;               // 4*14 = 56

  // ---- workspace layout (fp32) ----
  float* W = (float*)d_ws;
  size_t o = 0;
  float* X   = W + o; o += (size_t)Nn * HIDD;
  float* Y   = W + o; o += (size_t)Ne * HIDD;
  float* Z   = W + o; o += (size_t)Nt * HIDD;
  float* ME  = W + o; o += (size_t)Ne * HIDD;   // node-eggc messages
  float* MT  = W + o; o += (size_t)Nt * HIDD;   // edge-eggc messages
  float* SG  = W + o; o += (size_t)Ne * HIDD;   // also reused as mlp hidden
  float* DG  = W + o; o += (size_t)Ne * HIDD;
  float* BH  = W + o; o += (size_t)Ne * HIDD;
  float* SSH = W + o; o += (size_t)Ne * HIDD;
  float* SS  = W + o; o += (size_t)Ne * HIDD;
  float* ST  = W + o; o += 1024;
  float* GF  = W + o; o += (size_t)NBg * HIDD;
  float* CNT = W + o; o += NBg;

  // mlp: out = silu(bn(in @ w + b))   (leaf order: bn.b, bn.g, lin.b, lin.w)
  auto mlp = [&](const float* in, int M, int K, int Nc, int L, float* out) {
    h_gemm(in, P(L + 3), P(L + 2), out, M, K, Nc, 0, stream);
    h_stats(out, M, Nc, ST, stream);
    h_bnsilu(out, ST, Nc, P(L + 1), P(L + 0), out, M, 0, stream);
  };

  // eggc in-place: Xf (Rn x 256) and Ef (Re x 256) both updated
  auto eggc = [&](int L, float* Xf, float* Ef, float* msg,
                  const int* si, const int* di, int Rn, int Re) {
    const float *bnEb = P(L+0), *bnEg = P(L+1), *bnNb = P(L+2), *bnNg = P(L+3);
    const float *dgb = P(L+4), *dgw = P(L+5), *dub = P(L+6), *duw = P(L+7);
    const float *egb = P(L+8), *egw = P(L+9), *sgb = P(L+10), *sgw = P(L+11);
    const float *sub = P(L+12), *suw = P(L+13);

    h_gemm(Xf, sgw, sgb, SG, Rn, 256, 256, 0, stream);
    h_gemm(Xf, dgw, dgb, DG, Rn, 256, 256, 0, stream);
    h_gemm(Xf, duw, dub, BH, Rn, 256, 256, 0, stream);
    h_gemm(Ef, egw, egb, msg, Re, 256, 256, 0, stream);   // msg = edge_gate(e)+b

    long long nseg = (long long)Rn * 256;
    h_zero(SSH, nseg, stream);
    h_zero(SS, nseg, stream);
    long long nt = (long long)Re * 64;
    k_scatter<<<cdivu(nt, 256), 256, 0, stream>>>(
        (const float4*)SG, (const float4*)DG, (const float4*)BH, (float4*)msg,
        SSH, SS, si, di, Re);
    k_hdiv<<<cdivu(nseg, 256), 256, 0, stream>>>(SSH, SS, nseg);

    h_gemm(Xf, suw, sub, SSH, Rn, 256, 256, 1, stream);   // SSH = src_update(x)+h
    h_stats(SSH, Rn, 256, ST, stream);
    h_bnsilu(SSH, ST, 256, bnNg, bnNb, Xf, Rn, 1, stream); // x += silu(bn(.))
    h_stats(msg, Re, 256, ST, stream);
    h_bnsilu(msg, ST, 256, bnEg, bnEb, Ef, Re, 1, stream); // e += silu(bn(m))
  };

  // ---- embeddings ----
  mlp(x_atom, Nn, 92, 256, L_ATOM, X);
  float* H = SG;                                // scratch hidden buffer
  mlp(e_edge, Ne, 80, 64, L_EDGEE, H);
  mlp(H, Ne, 64, 256, L_EDGEE + 4, Y);
  mlp(t_trip, Nt, 40, 64, L_ANGLE, H);          // T*64 fits in SG (E*256)
  mlp(H, Nt, 64, 256, L_ANGLE + 4, Z);

  // ---- 4 ALIGNN layers: node eggc then edge eggc ----
  for (int i = 0; i < 4; ++i) {
    int Le = L_ALIGNN + i * 28;       // edge eggc leaves
    int Lnode = Le + 14;              // node eggc leaves
    eggc(Lnode, X, Y, ME, src, dst, Nn, Ne);
    eggc(Le, Y, Z, MT, lg_src, lg_dst, Ne, Nt);
  }
  // ---- 4 GCN layers ----
  for (int i = 0; i < 4; ++i)
    eggc(L_GCN + i * 14, X, Y, ME, src, dst, Nn, Ne);

  // ---- graph mean pooling ----
  h_zero(GF, (long long)NBg * 256 + NBg, stream);   // GF and CNT are contiguous
  k_poolacc<<<cdivu((long long)Nn * 64, 256), 256, 0, stream>>>(
      (const float4*)X, n2g, GF, CNT, Nn);
  k_poolfin<<<NBg, 256, 0, stream>>>(GF, CNT, NBg);

  // ---- attention (seq-1 collapse) + LN + fc ----
  // attn leaves: g2t_k(b,w)=133,134  g2t_q=135,136  g2t_v=137,138
  //              graph_out=139,140  ln_g(b,g)=141,142
  k_finale<<<NBg, 256, 0, stream>>>(
      GF, text,
      P(L_ATTN + 5), P(L_ATTN + 4),      // g2t_v.w, g2t_v.b
      P(L_ATTN + 7), P(L_ATTN + 6),      // graph_out.w, graph_out.b
      P(L_ATTN + 9), P(L_ATTN + 8),      // ln_g.g, ln_g.b
      P(L_FC + 1), P(L_FC + 0),          // fc.w, fc.b
      (float*)d_out);
}